// IMcgcnn_11458972746250
// MI455X (gfx1250) — compile-verified
//
#include <hip/hip_runtime.h>
#include <hip/hip_bf16.h>

#define N_NODES 40000
#define N_EDGES 250000
#define N_GRAPHS 512
#define H 128
#define IN_C 92
#define E_DIM 41
#define NLAYERS 3
#define EPS_BN 1e-5f
#define TILE_M 16      // row tile for generic GEMM
#define TILE_ME 32     // edge tile for edge kernels (z = 32x384 f32 = 48KB LDS)

typedef __attribute__((ext_vector_type(2))) float v2f;
typedef __attribute__((ext_vector_type(8))) float v8f;
typedef __attribute__((ext_vector_type(4))) int   v4i;

#if defined(__has_builtin)
#if __has_builtin(__builtin_amdgcn_global_load_async_to_lds_b128)
#define HAVE_ASYNC_LDS 1
#endif
#endif

#if defined(HAVE_ASYNC_LDS)
__device__ inline void cp16_async(void* lds, const void* g) {
  __builtin_amdgcn_global_load_async_to_lds_b128(
      (__attribute__((address_space(1))) v4i*)g,
      (__attribute__((address_space(3))) v4i*)lds, 0, 0);
}
#if __has_builtin(__builtin_amdgcn_s_wait_asynccnt)
#define WAIT_ASYNC() __builtin_amdgcn_s_wait_asynccnt(0)
#else
#define WAIT_ASYNC() asm volatile("s_wait_asynccnt 0x0" ::: "memory")
#endif
#else
#define WAIT_ASYNC()
#endif

// D = A(16x4 f32) * B(4x16 f32) + C(16x16 f32). fp32 WMMA: the problem is
// memory/gather bound (~220 GFLOP vs multi-TB/s of traffic), so full precision
// on the matrix path costs nothing.
__device__ inline v8f wmma4(v2f a, v2f b, v8f c) {
  return __builtin_amdgcn_wmma_f32_16x16x4_f32(
      false, a, false, b, (short)0, c, false, false);
}

__device__ inline float sigmoidf_(float x) { return 1.0f / (1.0f + expf(-x)); }
__device__ inline float softplusf_(float x) {
  return (x > 20.0f) ? x : log1pf(expf(x));
}

// ---------------------------------------------------------------------------
// Generic GEMM: Y[M,128] = act(X[M,Kreal] @ W[Kreal,128] + bias)
// ---------------------------------------------------------------------------
__global__ void __launch_bounds__(128)
gemm_bias_act(const float* __restrict__ X, const float* __restrict__ W,
              const float* __restrict__ bias, float* __restrict__ Y,
              int M, int Kreal, int Kpad, int act) {
  __shared__ float xs[TILE_M][128];
  const int tid = threadIdx.x;
  const int m0 = blockIdx.x * TILE_M;

  if (Kreal == Kpad && (Kreal & 3) == 0) {
    const int nq = Kpad >> 2;
    for (int i = tid; i < TILE_M * nq; i += 128) {
      int r = i / nq, q = i - r * nq;
      int row = m0 + r;
      float4 v = {0.f, 0.f, 0.f, 0.f};
      if (row < M) v = *(const float4*)&X[(size_t)row * Kreal + q * 4];
      *(float4*)&xs[r][q * 4] = v;
    }
  } else {
    for (int i = tid; i < TILE_M * Kpad; i += 128) {
      int r = i / Kpad, c = i - r * Kpad;
      int row = m0 + r;
      xs[r][c] = (row < M && c < Kreal) ? X[(size_t)row * Kreal + c] : 0.0f;
    }
  }
  __syncthreads();

  const int lane = tid & 31;
  const int wave = tid >> 5;
  const int hi = lane >> 4;
  const int ml = lane & 15;
  const int c0 = wave * 32 + ml;
  const int c1 = c0 + 16;

  v8f acc0 = {}, acc1 = {};
  for (int k = 0; k < Kpad; k += 4) {
    if ((k & 31) == 0 && (k + 32) < Kpad)
      __builtin_prefetch(&W[(k + 32) * H + c0], 0, 0);
    int kk = k + 2 * hi;
    v2f a;  a.x = xs[ml][kk];         a.y = xs[ml][kk + 1];
    v2f b0; b0.x = W[kk * H + c0];    b0.y = W[(kk + 1) * H + c0];
    v2f b1; b1.x = W[kk * H + c1];    b1.y = W[(kk + 1) * H + c1];
    acc0 = wmma4(a, b0, acc0);
    acc1 = wmma4(a, b1, acc1);
  }

#pragma unroll
  for (int r = 0; r < 8; ++r) {
    int m = m0 + r + 8 * hi;
    if (m < M) {
      float v0 = acc0[r] + bias[c0];
      float v1 = acc1[r] + bias[c1];
      if (act) { v0 = fmaxf(v0, 0.0f); v1 = fmaxf(v1, 0.0f); }
      Y[(size_t)m * H + c0] = v0;
      Y[(size_t)m * H + c1] = v1;
    }
  }
}

// ---------------------------------------------------------------------------
// Gather z = [h[dst], h[src], ea] for a TILE_ME-edge tile into LDS (16B chunks,
// async-to-LDS when available).
// ---------------------------------------------------------------------------
__device__ inline void gather_z(float (*z)[3 * H], int* dl, int* sl,
                                const float* __restrict__ h,
                                const float* __restrict__ ea,
                                const int* __restrict__ ei, int e0) {
  const int tid = threadIdx.x;
  if (tid < TILE_ME) {
    int e = e0 + tid;
    dl[tid] = (e < N_EDGES) ? ei[N_EDGES + e] : 0;   // edge_index[1] = dst
  } else if (tid < 2 * TILE_ME) {
    int e = e0 + tid - TILE_ME;
    sl[tid - TILE_ME] = (e < N_EDGES) ? ei[e] : 0;   // edge_index[0] = src
  }
  __syncthreads();
  // 96 float4 chunks per edge row: [h[dst] | h[src] | ea]
  for (int i = tid; i < TILE_ME * 96; i += 128) {
    int r = i / 96, q = i - r * 96;
    int e = e0 + r;
    float* lp = &z[r][q * 4];
    if (e < N_EDGES) {
      const float* gp;
      if (q < 32)      gp = h  + (size_t)dl[r] * H + q * 4;
      else if (q < 64) gp = h  + (size_t)sl[r] * H + (q - 32) * 4;
      else             gp = ea + (size_t)e     * H + (q - 64) * 4;
#if defined(HAVE_ASYNC_LDS)
      cp16_async(lp, gp);
#else
      *(float4*)lp = *(const float4*)gp;
#endif
    } else {
      float4 zf = {0.f, 0.f, 0.f, 0.f};
      *(float4*)lp = zf;
    }
  }
  WAIT_ASYNC();
  __syncthreads();
}

// ---------------------------------------------------------------------------
// Edge message: msg = softplus(z@Wsm+bsm) * sigmoid(z@Wcv+bcv); scatter-add.
// 32 edges/block; per wave: 2 M-tiles x 2 N-tiles x 2 matrices = 8 WMMA chains.
// ---------------------------------------------------------------------------
__global__ void __launch_bounds__(128)
edge_msg_kernel(const float* __restrict__ h, const float* __restrict__ ea,
                const int* __restrict__ ei, const float* __restrict__ Wsm,
                const float* __restrict__ bsm, const float* __restrict__ Wcv,
                const float* __restrict__ bcv, float* __restrict__ agg) {
  __shared__ float z[TILE_ME][3 * H];
  __shared__ int dl[TILE_ME], sl[TILE_ME];
  const int e0 = blockIdx.x * TILE_ME;
  gather_z(z, dl, sl, h, ea, ei, e0);

  const int lane = threadIdx.x & 31;
  const int wave = threadIdx.x >> 5;
  const int hi = lane >> 4;
  const int ml = lane & 15;
  const int c0 = wave * 32 + ml;
  const int c1 = c0 + 16;

  v8f aS0[2] = {}, aS1[2] = {}, aC0[2] = {}, aC1[2] = {};
  for (int k = 0; k < 3 * H; k += 4) {
    if ((k & 31) == 0 && (k + 32) < 3 * H) {
      __builtin_prefetch(&Wsm[(k + 32) * H + c0], 0, 0);
      __builtin_prefetch(&Wcv[(k + 32) * H + c0], 0, 0);
    }
    int kk = k + 2 * hi;
    v2f bs0; bs0.x = Wsm[kk * H + c0];  bs0.y = Wsm[(kk + 1) * H + c0];
    v2f bs1; bs1.x = Wsm[kk * H + c1];  bs1.y = Wsm[(kk + 1) * H + c1];
    v2f bc0; bc0.x = Wcv[kk * H + c0];  bc0.y = Wcv[(kk + 1) * H + c0];
    v2f bc1; bc1.x = Wcv[kk * H + c1];  bc1.y = Wcv[(kk + 1) * H + c1];
#pragma unroll
    for (int mt = 0; mt < 2; ++mt) {
      v2f a; a.x = z[mt * 16 + ml][kk]; a.y = z[mt * 16 + ml][kk + 1];
      aS0[mt] = wmma4(a, bs0, aS0[mt]);
      aS1[mt] = wmma4(a, bs1, aS1[mt]);
      aC0[mt] = wmma4(a, bc0, aC0[mt]);
      aC1[mt] = wmma4(a, bc1, aC1[mt]);
    }
  }

  const float bS0 = bsm[c0], bS1 = bsm[c1], bC0 = bcv[c0], bC1 = bcv[c1];
#pragma unroll
  for (int mt = 0; mt < 2; ++mt) {
#pragma unroll
    for (int r = 0; r < 8; ++r) {
      int m = mt * 16 + r + 8 * hi;
      int e = e0 + m;
      if (e < N_EDGES) {
        float v0 = softplusf_(aS0[mt][r] + bS0) * sigmoidf_(aC0[mt][r] + bC0);
        float v1 = softplusf_(aS1[mt][r] + bS1) * sigmoidf_(aC1[mt][r] + bC1);
        size_t rowo = (size_t)dl[m] * H;
        unsafeAtomicAdd(&agg[rowo + c0], v0);
        unsafeAtomicAdd(&agg[rowo + c1], v1);
      }
    }
  }
}

// ---------------------------------------------------------------------------
// Edge update pre-BN: t = sigmoid(z2 @ We + be). BN channel stats (sum, sumsq)
// are fused into the epilogue: register accumulate -> shfl_xor(16) -> atomics.
// ---------------------------------------------------------------------------
__global__ void __launch_bounds__(128)
edge_update_kernel(const float* __restrict__ h, const float* __restrict__ ea,
                   const int* __restrict__ ei, const float* __restrict__ We,
                   const float* __restrict__ be, float* __restrict__ t,
                   float* __restrict__ sums, float* __restrict__ sumsq) {
  __shared__ float z[TILE_ME][3 * H];
  __shared__ int dl[TILE_ME], sl[TILE_ME];
  const int e0 = blockIdx.x * TILE_ME;
  gather_z(z, dl, sl, h, ea, ei, e0);

  const int lane = threadIdx.x & 31;
  const int wave = threadIdx.x >> 5;
  const int hi = lane >> 4;
  const int ml = lane & 15;
  const int c0 = wave * 32 + ml;
  const int c1 = c0 + 16;

  v8f a0[2] = {}, a1[2] = {};
  for (int k = 0; k < 3 * H; k += 4) {
    if ((k & 31) == 0 && (k + 32) < 3 * H)
      __builtin_prefetch(&We[(k + 32) * H + c0], 0, 0);
    int kk = k + 2 * hi;
    v2f b0; b0.x = We[kk * H + c0]; b0.y = We[(kk + 1) * H + c0];
    v2f b1; b1.x = We[kk * H + c1]; b1.y = We[(kk + 1) * H + c1];
#pragma unroll
    for (int mt = 0; mt < 2; ++mt) {
      v2f a; a.x = z[mt * 16 + ml][kk]; a.y = z[mt * 16 + ml][kk + 1];
      a0[mt] = wmma4(a, b0, a0[mt]);
      a1[mt] = wmma4(a, b1, a1[mt]);
    }
  }

  const float bE0 = be[c0], bE1 = be[c1];
  float s0 = 0.f, q0 = 0.f, s1 = 0.f, q1 = 0.f;
#pragma unroll
  for (int mt = 0; mt < 2; ++mt) {
#pragma unroll
    for (int r = 0; r < 8; ++r) {
      int m = mt * 16 + r + 8 * hi;
      int e = e0 + m;
      if (e < N_EDGES) {
        float v0 = sigmoidf_(a0[mt][r] + bE0);
        float v1 = sigmoidf_(a1[mt][r] + bE1);
        t[(size_t)e * H + c0] = v0;
        t[(size_t)e * H + c1] = v1;
        s0 += v0; q0 += v0 * v0;
        s1 += v1; q1 += v1 * v1;
      }
    }
  }
  // combine the two half-rows (lanes l and l+16 hold complementary M rows)
  s0 += __shfl_xor(s0, 16, 32);
  q0 += __shfl_xor(q0, 16, 32);
  s1 += __shfl_xor(s1, 16, 32);
  q1 += __shfl_xor(q1, 16, 32);
  if (hi == 0) {
    unsafeAtomicAdd(&sums[c0], s0);
    unsafeAtomicAdd(&sumsq[c0], q0);
    unsafeAtomicAdd(&sums[c1], s1);
    unsafeAtomicAdd(&sumsq[c1], q1);
  }
}

// ---------------------------------------------------------------------------
// BatchNorm stats (float4): per-channel sum & sumsq, chunked + atomics.
// ---------------------------------------------------------------------------
#define STAT_ROWS 1024
__global__ void __launch_bounds__(256)
stats_partial(const float* __restrict__ X, int M, float* __restrict__ sums,
              float* __restrict__ sumsq) {
  __shared__ float4 ss[256], sq[256];
  const int tid = threadIdx.x;
  const int cg = tid & 31;    // channels 4*cg .. 4*cg+3
  const int ph = tid >> 5;    // row phase 0..7
  int rend = blockIdx.x * STAT_ROWS + STAT_ROWS;
  if (rend > M) rend = M;
  float4 s = {0.f, 0.f, 0.f, 0.f}, q = {0.f, 0.f, 0.f, 0.f};
  for (int r = blockIdx.x * STAT_ROWS + ph; r < rend; r += 8) {
    float4 v = *(const float4*)&X[(size_t)r * H + 4 * cg];
    s.x += v.x; s.y += v.y; s.z += v.z; s.w += v.w;
    q.x += v.x * v.x; q.y += v.y * v.y; q.z += v.z * v.z; q.w += v.w * v.w;
  }
  ss[tid] = s; sq[tid] = q;
  __syncthreads();
  if (tid < 32) {
    float4 S = ss[tid], Q = sq[tid];
#pragma unroll
    for (int p = 1; p < 8; ++p) {
      float4 a = ss[tid + 32 * p], b = sq[tid + 32 * p];
      S.x += a.x; S.y += a.y; S.z += a.z; S.w += a.w;
      Q.x += b.x; Q.y += b.y; Q.z += b.z; Q.w += b.w;
    }
    unsafeAtomicAdd(&sums[4 * tid + 0], S.x);
    unsafeAtomicAdd(&sums[4 * tid + 1], S.y);
    unsafeAtomicAdd(&sums[4 * tid + 2], S.z);
    unsafeAtomicAdd(&sums[4 * tid + 3], S.w);
    unsafeAtomicAdd(&sumsq[4 * tid + 0], Q.x);
    unsafeAtomicAdd(&sumsq[4 * tid + 1], Q.y);
    unsafeAtomicAdd(&sumsq[4 * tid + 2], Q.z);
    unsafeAtomicAdd(&sumsq[4 * tid + 3], Q.w);
  }
}

__global__ void finalize_stats(const float* __restrict__ sums,
                               const float* __restrict__ sumsq, float invM,
                               float* __restrict__ mu, float* __restrict__ rsig) {
  int c = threadIdx.x;
  if (c < H) {
    float m = sums[c] * invM;
    float v = sumsq[c] * invM - m * m;
    mu[c] = m;
    rsig[c] = rsqrtf(fmaxf(v, 0.0f) + EPS_BN);
  }
}

// dst[i] = (src[i]-mu)*rsig*gamma + beta + dst[i]   (BN + residual, float4)
__global__ void __launch_bounds__(256)
bn_residual(const float* __restrict__ src, const float* __restrict__ mu,
            const float* __restrict__ rsig, const float* __restrict__ gamma,
            const float* __restrict__ beta, float* __restrict__ dst, long n4) {
  long i = (long)blockIdx.x * blockDim.x + threadIdx.x;
  if (i < n4) {
    size_t base = (size_t)i * 4;
    int c = (int)(base & (H - 1));
    float4 s = *(const float4*)&src[base];
    float4 d = *(const float4*)&dst[base];
    d.x += (s.x - mu[c + 0]) * rsig[c + 0] * gamma[c + 0] + beta[c + 0];
    d.y += (s.y - mu[c + 1]) * rsig[c + 1] * gamma[c + 1] + beta[c + 1];
    d.z += (s.z - mu[c + 2]) * rsig[c + 2] * gamma[c + 2] + beta[c + 2];
    d.w += (s.w - mu[c + 3]) * rsig[c + 3] * gamma[c + 3] + beta[c + 3];
    *(float4*)&dst[base] = d;
  }
}

// ---------------------------------------------------------------------------
// Pooling + head
// ---------------------------------------------------------------------------
__global__ void __launch_bounds__(256)
pool_scatter(const float* __restrict__ p, const int* __restrict__ batch,
             float* __restrict__ gsum, float* __restrict__ gcnt) {
  long i = (long)blockIdx.x * blockDim.x + threadIdx.x;   // N_NODES * 32 chunks
  if (i < (long)N_NODES * 32) {
    int node = (int)(i >> 5);
    int qc = (int)(i & 31) * 4;
    int b = batch[node];
    float4 v = *(const float4*)&p[(size_t)node * H + qc];
    unsafeAtomicAdd(&gsum[(size_t)b * H + qc + 0], v.x);
    unsafeAtomicAdd(&gsum[(size_t)b * H + qc + 1], v.y);
    unsafeAtomicAdd(&gsum[(size_t)b * H + qc + 2], v.z);
    unsafeAtomicAdd(&gsum[(size_t)b * H + qc + 3], v.w);
    if (qc == 0) unsafeAtomicAdd(&gcnt[b], 1.0f);
  }
}

__global__ void graph_mean(const float* __restrict__ gsum,
                           const float* __restrict__ gcnt, float* __restrict__ g) {
  int i = blockIdx.x * blockDim.x + threadIdx.x;
  if (i < N_GRAPHS * H) g[i] = gsum[i] / fmaxf(gcnt[i >> 7], 1.0f);
}

__global__ void out_head(const float* __restrict__ g, const float* __restrict__ Wout,
                         const float* __restrict__ bout, float* __restrict__ out) {
  int gi = blockIdx.x * blockDim.x + threadIdx.x;
  if (gi < N_GRAPHS) {
    float acc = bout[0];
#pragma unroll 8
    for (int c = 0; c < H; ++c) acc += g[(size_t)gi * H + c] * Wout[c];
    out[gi] = acc;
  }
}

// ---------------------------------------------------------------------------
extern "C" void kernel_launch(void* const* d_in, const int* in_sizes, int n_in,
                              void* d_out, int out_size, void* d_ws, size_t ws_size,
                              hipStream_t stream) {
  const float* x         = (const float*)d_in[0];
  const float* edge_attr = (const float*)d_in[1];
  const int*   ei        = (const int*)d_in[2];
  const int*   batch     = (const int*)d_in[3];
  const float* Wpre_n = (const float*)d_in[4];
  const float* bpre_n = (const float*)d_in[5];
  const float* Wpre_e = (const float*)d_in[6];
  const float* bpre_e = (const float*)d_in[7];
  const float* Wsm = (const float*)d_in[8];
  const float* bsm = (const float*)d_in[9];
  const float* Wcv = (const float*)d_in[10];
  const float* bcv = (const float*)d_in[11];
  const float* We  = (const float*)d_in[12];
  const float* be  = (const float*)d_in[13];
  const float* g_node = (const float*)d_in[14];
  const float* b_node = (const float*)d_in[15];
  const float* g_edge = (const float*)d_in[16];
  const float* b_edge = (const float*)d_in[17];
  const float* Wpool = (const float*)d_in[18];
  const float* bpool = (const float*)d_in[19];
  const float* Wh  = (const float*)d_in[20];
  const float* bh  = (const float*)d_in[21];
  const float* Wout = (const float*)d_in[22];
  const float* bout = (const float*)d_in[23];
  float* out = (float*)d_out;

  float* ws = (float*)d_ws;
  size_t off = 0;
  float* h    = ws + off; off += (size_t)N_NODES * H;
  float* agg  = ws + off; off += (size_t)N_NODES * H;   // reused as pooled p
  float* ea   = ws + off; off += (size_t)N_EDGES * H;
  float* t    = ws + off; off += (size_t)N_EDGES * H;
  float* sums = ws + off; off += H;
  float* sumsq= ws + off; off += H;                     // contiguous with sums
  float* mu   = ws + off; off += H;
  float* rsig = ws + off; off += H;
  float* gsum = ws + off; off += (size_t)N_GRAPHS * H;
  float* gcnt = ws + off; off += N_GRAPHS;              // contiguous with gsum
  float* gA   = ws + off; off += (size_t)N_GRAPHS * H;
  float* gB   = ws + off; off += (size_t)N_GRAPHS * H;

  const int EB16 = (N_EDGES + TILE_M - 1) / TILE_M;     // 15625
  const int EB32 = (N_EDGES + TILE_ME - 1) / TILE_ME;   // 7813
  const int NB16 = (N_NODES + TILE_M - 1) / TILE_M;     // 2500
  const int GB16 = (N_GRAPHS + TILE_M - 1) / TILE_M;    // 32

  // pre-conv embeddings
  gemm_bias_act<<<NB16, 128, 0, stream>>>(x, Wpre_n, bpre_n, h, N_NODES, IN_C, IN_C, 1);
  gemm_bias_act<<<EB16, 128, 0, stream>>>(edge_attr, Wpre_e, bpre_e, ea, N_EDGES, E_DIM, 44, 1);

  for (int l = 0; l < NLAYERS; ++l) {
    const size_t wo = (size_t)l * 3 * H * H;
    // NodeConv
    (void)hipMemsetAsync(agg, 0, (size_t)N_NODES * H * sizeof(float), stream);
    edge_msg_kernel<<<EB32, 128, 0, stream>>>(h, ea, ei, Wsm + wo, bsm + l * H,
                                              Wcv + wo, bcv + l * H, agg);
    (void)hipMemsetAsync(sums, 0, 2 * H * sizeof(float), stream);
    stats_partial<<<(N_NODES + STAT_ROWS - 1) / STAT_ROWS, 256, 0, stream>>>(
        agg, N_NODES, sums, sumsq);
    finalize_stats<<<1, H, 0, stream>>>(sums, sumsq, 1.0f / N_NODES, mu, rsig);
    bn_residual<<<((long)N_NODES * H / 4 + 255) / 256, 256, 0, stream>>>(
        agg, mu, rsig, g_node + l * H, b_node + l * H, h, (long)N_NODES * H / 4);
    // EdgeUpdate (BN stats fused into epilogue)
    (void)hipMemsetAsync(sums, 0, 2 * H * sizeof(float), stream);
    edge_update_kernel<<<EB32, 128, 0, stream>>>(h, ea, ei, We + wo, be + l * H,
                                                 t, sums, sumsq);
    finalize_stats<<<1, H, 0, stream>>>(sums, sumsq, 1.0f / N_EDGES, mu, rsig);
    bn_residual<<<((long)N_EDGES * H / 4 + 255) / 256, 256, 0, stream>>>(
        t, mu, rsig, g_edge + l * H, b_edge + l * H, ea, (long)N_EDGES * H / 4);
  }

  // pre-pool projection + mean pool
  gemm_bias_act<<<NB16, 128, 0, stream>>>(h, Wpool, bpool, agg, N_NODES, H, H, 1);
  (void)hipMemsetAsync(gsum, 0, ((size_t)N_GRAPHS * H + N_GRAPHS) * sizeof(float), stream);
  pool_scatter<<<((long)N_NODES * 32 + 255) / 256, 256, 0, stream>>>(agg, batch, gsum, gcnt);
  graph_mean<<<(N_GRAPHS * H + 255) / 256, 256, 0, stream>>>(gsum, gcnt, gA);

  // head
  gemm_bias_act<<<GB16, 128, 0, stream>>>(gA, Wh, bh, gB, N_GRAPHS, H, H, 1);
  gemm_bias_act<<<GB16, 128, 0, stream>>>(gB, Wh + H * H, bh + H, gA, N_GRAPHS, H, H, 1);
  out_head<<<(N_GRAPHS + 255) / 256, 256, 0, stream>>>(gA, Wout, bout, out);
}